// BCNLayer_39745627357597
// MI455X (gfx1250) — compile-verified
//
#include <hip/hip_runtime.h>
#include <cstdint>

// Problem constants (match reference)
#define HH   128
#define WW   128
#define HWSZ (HH * WW)      // 16384
#define BB   4096

// Tiling
#define TY    4             // output rows per block
#define TX    64            // output cols per block
#define TB    32            // batch floats per block (one 128B cacheline per position)
#define TROWS (TY + 2)      // 6  halo rows
#define TCOLS (TX + 2)      // 66 halo cols
#define NPOS  (TROWS * TCOLS) // 396 staged positions
#define SLOT  36            // padded floats per position (32 data + 4 pad) -> conflict-free b128 reads
#define LDS_FLOATS (NPOS * SLOT) // 14256 floats = 55.7 KB

__device__ __forceinline__ float sigf(float v) {
    // v_exp_f32 + v_rcp_f32 (TRANS ops) instead of the IEEE divide expansion
    return __builtin_amdgcn_rcpf(1.0f + __expf(-v));
}

__device__ __forceinline__ int clampi(int v, int lo, int hi) {
    return v < lo ? lo : (v > hi ? hi : v);
}

__global__ __launch_bounds__(256) void bcn_stencil_kernel(
        const float* __restrict__ x,   // (HW, B)
        const float* __restrict__ w,   // (3, 3, HW)
        float* __restrict__ out)       // (HW, B)
{
    __shared__ float tile[LDS_FLOATS];

    const int t  = threadIdx.x;
    const int x0 = blockIdx.x * TX;
    const int y0 = blockIdx.y * TY;
    const int b0 = blockIdx.z * TB;

    const int q  = t & 3;        // batch quarter (f4 index 0..3; second pass adds +4)
    const int px = t >> 2;       // 0..63 output column within tile

    // ---------------- Preload per-column weights FIRST (overlaps async staging) ----------
    // Output (py, px); tap with tile row ry = py+r3 and col offset c:
    //   weight plane = (2-r3)*3 + (2-c), sampled at the SOURCE position
    //   (y0+py+r3-1, x0+px+c-1); out-of-grid source -> weight forced to 0 (matches zero pad).
    // Branchless: clamp coords so the load is always legal, then cndmask the value.
    float wv[TY][3][3];
#pragma unroll
    for (int py = 0; py < TY; ++py) {
#pragma unroll
        for (int r3 = 0; r3 < 3; ++r3) {
#pragma unroll
            for (int c = 0; c < 3; ++c) {
                const int ys = y0 + py + r3 - 1;
                const int xs = x0 + px + c - 1;
                const bool ok = ((unsigned)ys < (unsigned)HH) && ((unsigned)xs < (unsigned)WW);
                const int ysc = clampi(ys, 0, HH - 1);
                const int xsc = clampi(xs, 0, WW - 1);
                const float v = w[((2 - r3) * 3 + (2 - c)) * HWSZ + ysc * WW + xsc];
                wv[py][r3][c] = ok ? v : 0.0f;
            }
        }
    }

    // ---------------- Stage x halo tile: global -> LDS via CDNA5 async copies -------------
    {
        const int      lid   = t & 7;              // which 16B chunk of a position's 128B line
        const uint64_t xbase = (uint64_t)(uintptr_t)x;
        // p0 = t>>3 in [0,32) < TCOLS, so ry0 = 0, rx0 = p0: no integer division needed.
        int ry = 0;
        int rx = t >> 3;
        uint32_t loff = (uint32_t)(uintptr_t)(&tile[(t >> 3) * SLOT + lid * 4]);
        const uint32_t lstep = 32u * SLOT * 4u;    // LDS bytes advanced per iteration
        for (int p = t >> 3; p < NPOS; p += 32) {
            const int ys = y0 + ry - 1;
            const int xs = x0 + rx - 1;
            if ((unsigned)ys < (unsigned)HH && (unsigned)xs < (unsigned)WW) {
                const uint32_t goff =
                    (uint32_t)((((ys * WW + xs) * (size_t)BB) + b0 + lid * 4) * 4u);
                // GLOBAL_LOAD_ASYNC_TO_LDS_B128  vdst(lds-addr), vaddr(offset), saddr(base)
                asm volatile("global_load_async_to_lds_b128 %0, %1, %2"
                             :: "v"(loff), "v"(goff), "s"(xbase)
                             : "memory");
            } else {
                // zero-fill out-of-grid halo (regular DS store, disjoint from async targets)
                *(float4*)(&tile[p * SLOT + lid * 4]) = make_float4(0.f, 0.f, 0.f, 0.f);
            }
            loff += lstep;
            rx += 32;
            if (rx >= TCOLS) { rx -= TCOLS; ++ry; }   // stride 32 < TCOLS: single wrap
        }
    }
    // drain this wave's async copies, then cross-wave barrier (syncthreads also waits dscnt)
#if __has_builtin(__builtin_amdgcn_s_wait_asynccnt)
    __builtin_amdgcn_s_wait_asynccnt(0);
#else
    asm volatile("s_wait_asynccnt 0" ::: "memory");
#endif
    __syncthreads();

    // ---------------- Compute ------------------------------------------------------------
#pragma unroll
    for (int kb = 0; kb < 2; ++kb) {
        const int kk = q + kb * 4;   // which f4 of the 32-float batch slice

        float4 acc[TY];
#pragma unroll
        for (int py = 0; py < TY; ++py) acc[py] = make_float4(0.f, 0.f, 0.f, 0.f);

        // Sliding-row accumulation: 6 rows x 3 cols of f4 loads feed 4 outputs
        // (4.5 LDS f4 loads per output instead of 9).
#pragma unroll
        for (int ry = 0; ry < TROWS; ++ry) {
            float4 xr[3];
#pragma unroll
            for (int c = 0; c < 3; ++c)
                xr[c] = *(const float4*)(&tile[(ry * TCOLS + px + c) * SLOT + kk * 4]);
#pragma unroll
            for (int py = 0; py < TY; ++py) {
                const int r3 = ry - py;
                if (r3 < 0 || r3 > 2) continue;   // constant-folds under full unroll
#pragma unroll
                for (int c = 0; c < 3; ++c) {
                    const float wt = wv[py][r3][c];
                    acc[py].x = fmaf(wt, xr[c].x, acc[py].x);
                    acc[py].y = fmaf(wt, xr[c].y, acc[py].y);
                    acc[py].z = fmaf(wt, xr[c].z, acc[py].z);
                    acc[py].w = fmaf(wt, xr[c].w, acc[py].w);
                }
            }
        }

#pragma unroll
        for (int py = 0; py < TY; ++py) {
            float4 r;
            r.x = sigf(acc[py].x);
            r.y = sigf(acc[py].y);
            r.z = sigf(acc[py].z);
            r.w = sigf(acc[py].w);
            float* dst = out + ((size_t)((y0 + py) * WW + (x0 + px)) * BB) + b0 + kk * 4;
            *(float4*)dst = r;
        }
    }
}

extern "C" void kernel_launch(void* const* d_in, const int* in_sizes, int n_in,
                              void* d_out, int out_size, void* d_ws, size_t ws_size,
                              hipStream_t stream) {
    const float* x = (const float*)d_in[0];   // (HW, B) f32
    const float* w = (const float*)d_in[1];   // (3, 3, HW) f32
    float* out     = (float*)d_out;           // (HW, B) f32

    dim3 grid(WW / TX, HH / TY, BB / TB);     // (2, 32, 128) = 8192 blocks
    bcn_stencil_kernel<<<grid, 256, 0, stream>>>(x, w, out);
}